// GCN_33560874451187
// MI455X (gfx1250) — compile-verified
//
#include <hip/hip_runtime.h>

// ---------------------------------------------------------------------------
// GCN (1 -> 10 -> 1 features) on N=500K nodes, E=16M edges.
// Algebraically folded so each edge pass moves ONE scalar:
//   u[i]  = dinv[i]*x[i];          p1[c] = dinv[c]*(sum_e u[row_e] + u[c])
//   s2[i] = sum_k relu(p1[i]*W1[k]+b1[k])*W2[k]   (dense stage -> WMMA)
//   v[i]  = dinv[i]*s2[i];         p2[c] = dinv[c]*(sum_e v[row_e] + v[c])
//   out   = clip(p2 + b2, -0.5, 9.5)
// Edge list streams at HBM rate; gather/scatter arrays (2 MB) are L2-resident.
// ---------------------------------------------------------------------------

typedef float v2f __attribute__((ext_vector_type(2)));
typedef float v8f __attribute__((ext_vector_type(8)));

__global__ void k_zero_deg(unsigned int* __restrict__ deg, int n) {
    int i = blockIdx.x * blockDim.x + threadIdx.x;
    if (i < n) deg[i] = 0u;
}

// in-degree count (weights are all 1.0); self-loop added later as +1
__global__ void k_deg(const long long* __restrict__ col,
                      unsigned int* __restrict__ deg, int e) {
    int i = blockIdx.x * blockDim.x + threadIdx.x;
    if (i < e) atomicAdd(&deg[(int)col[i]], 1u);
}

// dinv = rsqrt(deg+1); u = dinv*x; acc1 starts with the self-loop term u[i]
__global__ void k_node1(const float* __restrict__ x,
                        const unsigned int* __restrict__ deg,
                        float* __restrict__ dinv,
                        float* __restrict__ u,
                        float* __restrict__ acc1, int n) {
    int i = blockIdx.x * blockDim.x + threadIdx.x;
    if (i < n) {
        float d  = (float)(deg[i] + 1u);       // always >= 1
        float di = rsqrtf(d);
        dinv[i]  = di;
        float ui = di * x[i];
        u[i]     = ui;
        acc1[i]  = ui;                          // self-loop contribution
    }
}

// one scalar per edge: acc[col] += src[row]  (src/acc are L2-resident, 2 MB)
__global__ void k_edge_prop(const long long* __restrict__ row,
                            const long long* __restrict__ col,
                            const float* __restrict__ src,
                            float* __restrict__ acc, int e) {
    int i = blockIdx.x * blockDim.x + threadIdx.x;
    if (i < e) {
        int r = (int)row[i];
        int c = (int)col[i];
        atomicAdd(&acc[c], src[r]);             // global_atomic_add_f32, no-return
    }
}

// Dense node transform via V_WMMA_F32_16X16X4_F32.
// One wave handles 16 nodes: A = p1 in K=0 column, B = W1 in K=0 row,
// C = b1 broadcast. D tile: VGPR j holds row M=j (lanes 0-15) / M=j+8
// (lanes 16-31), column N = lane%16. relu, scale by W2[N], reduce over the
// 16-lane half -> s2 per node. In-place: acc1 is read, acc2(=v) is written.
__global__ void k_node2_wmma(const float* __restrict__ dinv,
                             float* __restrict__ acc,     // in: acc1, out: acc2 init
                             float* __restrict__ v,       // out: v = dinv*s2
                             const float* __restrict__ W1,
                             const float* __restrict__ b1,
                             const float* __restrict__ W2, int n) {
    int lane = threadIdx.x & 31;
    int wave = threadIdx.x >> 5;
    int base = (blockIdx.x * (blockDim.x >> 5) + wave) * 16;
    if (base >= n) return;                       // wave-uniform exit (EXEC stays full)

    int m    = lane & 15;                        // node-in-tile / column index
    int node = base + m;
    int idx  = node < n ? node : n - 1;
    float p1 = dinv[idx] * acc[idx];
    if (node >= n) p1 = 0.0f;

    // A (16x4 f32): VGPR0 lanes 0-15 = K=0 column; everything else zero.
    v2f a; a.x = (lane < 16) ? p1 : 0.0f; a.y = 0.0f;
    // B (4x16 f32): row K=0 = W1 (padded to 16); rows 1..3 zero.
    float w1v = (m < 10) ? W1[m] : 0.0f;
    v2f b; b.x = (lane < 16) ? w1v : 0.0f; b.y = 0.0f;
    // C (16x16 f32): bias b1[N] independent of row.
    float b1v = (m < 10) ? b1[m] : 0.0f;
    v8f c;
#pragma unroll
    for (int j = 0; j < 8; ++j) c[j] = b1v;

    // D[M,N] = p1[M]*W1[N] + b1[N]
    v8f d = __builtin_amdgcn_wmma_f32_16x16x4_f32(
        /*neg_a=*/false, a, /*neg_b=*/false, b,
        /*c_mod=*/(short)0, c, /*reuse_a=*/false, /*reuse_b=*/false);

    float w2v = (m < 10) ? W2[m] : 0.0f;
    float r[8];
#pragma unroll
    for (int j = 0; j < 8; ++j) {
        float t = fmaxf(d[j], 0.0f) * w2v;       // relu * W2[N]
        t += __shfl_xor(t, 1, 32);               // reduce within each 16-lane half
        t += __shfl_xor(t, 2, 32);
        t += __shfl_xor(t, 4, 32);
        t += __shfl_xor(t, 8, 32);
        r[j] = t;  // lanes 0-15: s2(row j); lanes 16-31: s2(row j+8)
    }

    int rowbase = base + ((lane >= 16) ? 8 : 0);
    if (lane == 0 || lane == 16) {
#pragma unroll
        for (int j = 0; j < 8; ++j) {
            int nd = rowbase + j;
            if (nd < n) {
                float vv = dinv[nd] * r[j];      // v = dinv * s2
                v[nd]   = vv;
                acc[nd] = vv;                    // acc2 init = self-loop term
            }
        }
    }
}

__global__ void k_out(const float* __restrict__ dinv,
                      const float* __restrict__ acc2,
                      const float* __restrict__ b2,
                      float* __restrict__ out, int n) {
    int i = blockIdx.x * blockDim.x + threadIdx.x;
    if (i < n) {
        float p2 = dinv[i] * acc2[i] + b2[0];
        out[i] = fminf(fmaxf(p2, -0.5f), 9.5f);
    }
}

extern "C" void kernel_launch(void* const* d_in, const int* in_sizes, int n_in,
                              void* d_out, int out_size, void* d_ws, size_t ws_size,
                              hipStream_t stream) {
    const float*     x   = (const float*)d_in[0];
    const long long* ei  = (const long long*)d_in[1];   // int64 [2,E]
    const float*     W1  = (const float*)d_in[2];
    const float*     b1  = (const float*)d_in[3];
    const float*     W2  = (const float*)d_in[4];
    const float*     b2  = (const float*)d_in[5];

    int n = in_sizes[0];          // x is [N,1]
    int e = in_sizes[1] / 2;      // edge_index is [2,E]
    const long long* row = ei;
    const long long* col = ei + e;

    // workspace: deg(u32 N) | dinv(f32 N) | u/v(f32 N) | acc1/acc2(f32 N) = 16N B
    unsigned int* deg  = (unsigned int*)d_ws;
    float*        dinv = (float*)(deg + n);
    float*        uv   = dinv + n;
    float*        acc  = uv + n;

    const int tb = 256;
    int gn = (n + tb - 1) / tb;
    int ge = (e + tb - 1) / tb;
    int gt = (n + 127) / 128;     // 8 waves/block * 16 nodes/wave

    k_zero_deg  <<<gn, tb, 0, stream>>>(deg, n);
    k_deg       <<<ge, tb, 0, stream>>>(col, deg, e);
    k_node1     <<<gn, tb, 0, stream>>>(x, deg, dinv, uv, acc, n);
    k_edge_prop <<<ge, tb, 0, stream>>>(row, col, uv, acc, e);       // layer-1 scatter
    k_node2_wmma<<<gt, tb, 0, stream>>>(dinv, acc, uv, W1, b1, W2, n);
    k_edge_prop <<<ge, tb, 0, stream>>>(row, col, uv, acc, e);       // layer-2 scatter
    k_out       <<<gn, tb, 0, stream>>>(dinv, acc, b2, (float*)d_out, n);
}